// GemmRSIntraNode_62002147885352
// MI455X (gfx1250) — compile-verified
//
#include <hip/hip_runtime.h>
#include <stdint.h>

// GEMM + reduce-scatter (reshape) for MI455X / gfx1250, wave32.
//   C[m,n] = sum_r sum_k In[r,m,k] * W[r,n,k]   (exact fp32, V_WMMA_F32_16X16X4_F32)
// Data path: GLOBAL_LOAD_ASYNC_TO_LDS_B128 (ASYNCcnt) -> double-buffered LDS
//            -> ds_load_b64 fragments -> WMMA.  HBM-bound (~512 MB / 23.3 TB/s).

typedef __attribute__((ext_vector_type(2))) float v2f;
typedef __attribute__((ext_vector_type(8))) float v8f;

#define WS   8
#define GM   8192
#define GN   4096
#define LK   1024

#define BM   256            // workgroup tile M
#define BN   128            // workgroup tile N
#define KC   32             // K-chunk staged in LDS per iteration
#define LDA  36             // padded LDS row stride in floats (bank-conflict-free)
#define NCHUNK ((WS * LK) / KC)   // 256 chunks; LK % KC == 0 so no rank straddle

__global__ __launch_bounds__(256) void gemm_rs_f32_wmma_async(
    const float* __restrict__ A,   // [WS, GM, LK], k-contiguous
    const float* __restrict__ B,   // [WS, GN, LK], k-contiguous
    float* __restrict__ C)         // [GM, GN] row-major (== [WS, GM/WS, GN] flat)
{
    __shared__ float sA[2][BM * LDA];   // 2 x 36864 B
    __shared__ float sB[2][BN * LDA];   // 2 x 18432 B

    const int tid  = threadIdx.x;
    const int lane = tid & 31;
    const int wave = tid >> 5;
    const int wm   = wave >> 1;        // 0..3
    const int wn   = wave & 1;         // 0..1

    const int m0blk = blockIdx.y * BM;
    const int n0blk = blockIdx.x * BN;

    // ---- per-thread async-copy addressing (constant across chunks) ----
    // idx = tid + 256*q ; row = idx>>3 ; seg = idx&7 ; each copy is 16 B (b128).
    const int rA = tid >> 3;           // base row, +32 per q
    const int sg = tid & 7;            // 128-byte row covered by 8 lanes
    uint32_t voffA[8], ldsA0[8];       // A: 256 rows * 128 B = 8 b128 per thread
    uint32_t voffB[4], ldsB0[4];       // B: 128 rows * 128 B = 4 b128 per thread
#pragma unroll
    for (int q = 0; q < 8; ++q) {
        const int row = rA + 32 * q;
        voffA[q] = (uint32_t)(row * (LK * 4) + sg * 16);
        ldsA0[q] = (uint32_t)(uintptr_t)&sA[0][row * LDA + sg * 4];
    }
#pragma unroll
    for (int q = 0; q < 4; ++q) {
        const int row = rA + 32 * q;
        voffB[q] = (uint32_t)(row * (LK * 4) + sg * 16);
        ldsB0[q] = (uint32_t)(uintptr_t)&sB[0][row * LDA + sg * 4];
    }

    // Issue one K-chunk's async copies into LDS buffer `buf`.
    auto issue = [&](int c, int buf) {
        const int r    = c >> 5;             // chunk / (LK/KC)
        const int kloc = (c & 31) * KC;
        const float* sbA = A + ((size_t)r * GM + m0blk) * LK + kloc;
        const float* sbB = B + ((size_t)r * GN + n0blk) * LK + kloc;
        const uint32_t bofA = (uint32_t)buf * (uint32_t)(BM * LDA * 4);
        const uint32_t bofB = (uint32_t)buf * (uint32_t)(BN * LDA * 4);
#pragma unroll
        for (int q = 0; q < 8; ++q)
            asm volatile("global_load_async_to_lds_b128 %0, %1, %2"
                         :: "v"(ldsA0[q] + bofA), "v"(voffA[q]), "s"(sbA)
                         : "memory");
#pragma unroll
        for (int q = 0; q < 4; ++q)
            asm volatile("global_load_async_to_lds_b128 %0, %1, %2"
                         :: "v"(ldsB0[q] + bofB), "v"(voffB[q]), "s"(sbB)
                         : "memory");
    };

    // ---- WMMA fragment addressing (V_WMMA_F32_16X16X4_F32) ----
    //   lanes 0-15 : row = lane,    K pair {k, k+1}
    //   lanes 16-31: row = lane-16, K pair {k+2, k+3}
    const int rowl = lane & 15;
    const int kk   = (lane >> 4) << 1;     // 0 or 2

    v8f acc[4][4];
#pragma unroll
    for (int i = 0; i < 4; ++i)
#pragma unroll
        for (int j = 0; j < 4; ++j)
            acc[i][j] = (v8f){0.f, 0.f, 0.f, 0.f, 0.f, 0.f, 0.f, 0.f};

    // ---- pipeline: prefetch chunk 0, then overlap issue(c+1) with compute(c) ----
    issue(0, 0);
    asm volatile("s_wait_asynccnt 0x0" ::: "memory");
    __syncthreads();

    for (int c = 0; c < NCHUNK; ++c) {
        const int buf = c & 1;
        if (c + 1 < NCHUNK)
            issue(c + 1, buf ^ 1);

        const float* aw = &sA[buf][(wm * 64 + rowl) * LDA + kk];
        const float* bw = &sB[buf][(wn * 64 + rowl) * LDA + kk];

#pragma unroll
        for (int k4 = 0; k4 < KC; k4 += 4) {
            v2f af[4], bf[4];
#pragma unroll
            for (int i = 0; i < 4; ++i)
                af[i] = *(const v2f*)(aw + i * 16 * LDA + k4);
#pragma unroll
            for (int j = 0; j < 4; ++j)
                bf[j] = *(const v2f*)(bw + j * 16 * LDA + k4);
#pragma unroll
            for (int i = 0; i < 4; ++i)
#pragma unroll
                for (int j = 0; j < 4; ++j)
                    acc[i][j] = __builtin_amdgcn_wmma_f32_16x16x4_f32(
                        false, af[i], false, bf[j],
                        (short)0, acc[i][j], false, false);
        }

        // Prefetch for chunk c+1 must have landed before the next iteration;
        // every wave waits on its own ASYNCcnt, then the workgroup barrier
        // makes completion visible to all consumers.
        asm volatile("s_wait_asynccnt 0x0" ::: "memory");
        __syncthreads();
    }

    // ---- store: C/D 16x16 f32 layout: VGPR p -> M = p + 8*(lane>=16), N = lane&15
    const int crow = (lane >> 4) << 3;
    const int ccol = lane & 15;
    const int m0 = m0blk + wm * 64;
    const int n0 = n0blk + wn * 64;
#pragma unroll
    for (int i = 0; i < 4; ++i) {
#pragma unroll
        for (int j = 0; j < 4; ++j) {
            float* cp = C + (size_t)(m0 + i * 16 + crow) * GN
                          + (size_t)(n0 + j * 16 + ccol);
#pragma unroll
            for (int p = 0; p < 8; ++p)
                cp[(size_t)p * GN] = acc[i][j][p];
        }
    }
}

extern "C" void kernel_launch(void* const* d_in, const int* in_sizes, int n_in,
                              void* d_out, int out_size, void* d_ws, size_t ws_size,
                              hipStream_t stream) {
    (void)in_sizes; (void)n_in; (void)out_size; (void)d_ws; (void)ws_size;
    const float* inp = (const float*)d_in[0];   // [WS, GM, LK]
    const float* wgt = (const float*)d_in[1];   // [WS, GN, LK]
    float*       out = (float*)d_out;           // [GM, GN] flat

    dim3 grid(GN / BN, GM / BM);   // (32, 32)
    dim3 block(256, 1, 1);         // 8 wave32s
    gemm_rs_f32_wmma_async<<<grid, block, 0, stream>>>(inp, wgt, out);
}